// RadialDescriptor_7249904796076
// MI455X (gfx1250) — compile-verified
//
#include <hip/hip_runtime.h>
#include <math.h>

#define N_ATOMS 20000
#define NN      64
#define N_TYPES 4
#define N_DESC  8
#define K_MAX   8
#define R_C     8.0f

#define ATOMS_PER_BLOCK 64
#define THREADS         128

typedef __attribute__((ext_vector_type(2))) float v2f;
typedef __attribute__((ext_vector_type(8))) float v8f;

// out[a][d] = sum_j mask * sum_k c_table[ta][tj][d][k] * f_k(r_aj)
//
// Phase 1: per-edge Chebyshev basis accumulated into per-atom, per-neighbor-type
//          buckets in LDS (fp32 atomics -> exact reordering of the ref sum).
//          F is stored in a swizzled layout so that each phase-2 lane's 16
//          values are contiguous:  s = 16*h + 2*p + low, where the logical
//          slot m = tj*8+k decomposes as m = 4*p + 2*h + low
//          (p = tj*2 + (k>>2), h = (k>>1)&1, low = k&1).
// Phase 2: out[a][0:8] = G[a][0:128] x Cmat[128][8] with K expanded over the
//          4 possible center-types; 32 chained V_WMMA_F32_16X16X4_F32 per
//          16-atom tile. A fragments come from 4 ds_load_b128 preloads +
//          register cndmask; all 32 B fragments are batch-preloaded.
__global__ __launch_bounds__(THREADS)
void radial_desc_wmma_kernel(const int*   __restrict__ types,
                             const float* __restrict__ positions,
                             const int*   __restrict__ nbrs,
                             const float* __restrict__ offsets,
                             const float* __restrict__ ctab,
                             float*       __restrict__ out)
{
    __shared__ float F[ATOMS_PER_BLOCK * 32];   // [atom_local][32 swizzled], 8 KB

    const int tid      = threadIdx.x;
    const int atomBase = blockIdx.x * ATOMS_PER_BLOCK;

    // ---- zero the accumulation buckets ----
    #pragma unroll
    for (int i = tid; i < ATOMS_PER_BLOCK * 32; i += THREADS) F[i] = 0.0f;
    __syncthreads();

    // ---- Phase 1: stream edges, fully coalesced (edge = iter*128 + tid) ----
    const float inv_rc = 1.0f / R_C;
    #pragma unroll 2
    for (int iter = 0; iter < (ATOMS_PER_BLOCK * NN) / THREADS; ++iter) {
        const int e     = iter * THREADS + tid;   // 0..4095 within block
        const int aLoc  = e >> 6;                 // e / NN
        const int j     = e & (NN - 1);
        const int aGlob = atomBase + aLoc;
        if (aGlob < N_ATOMS) {
            const int eg = aGlob * NN + j;        // = atomBase*NN + e -> coalesced
            const int nb = nbrs[eg];
            if (nb >= 0) {
                const float ox = offsets[eg * 3 + 0];
                const float oy = offsets[eg * 3 + 1];
                const float oz = offsets[eg * 3 + 2];
                const float dx = positions[nb * 3 + 0] + ox - positions[aGlob * 3 + 0];
                const float dy = positions[nb * 3 + 1] + oy - positions[aGlob * 3 + 1];
                const float dz = positions[nb * 3 + 2] + oz - positions[aGlob * 3 + 2];
                const float r  = sqrtf(dx * dx + dy * dy + dz * dz);

                const float fc  = (r < R_C) ? (0.5f * cosf((float)M_PI * r * inv_rc) + 0.5f)
                                            : 0.0f;
                const float xr  = r * inv_rc - 1.0f;
                const float x   = 2.0f * xr * xr - 1.0f;
                const float hfc = 0.5f * fc;

                const int tj = types[nb];
                float* dstF = &F[aLoc * 32 + tj * 4];

                // swizzled slot for basis index k: 16*((k>>1)&1) + 2*(k>>2) + (k&1)
                float fkm2 = 1.0f;       // T0
                float fkm1 = x;          // T1
                float fk;
                atomicAdd(&dstF[0],  (1.0f + 1.0f) * hfc);                 // k=0 -> 0
                atomicAdd(&dstF[1],  (x    + 1.0f) * hfc);                 // k=1 -> 1
                fk = 2.0f * x * fkm1 - fkm2; fkm2 = fkm1; fkm1 = fk;
                atomicAdd(&dstF[16], (fk + 1.0f) * hfc);                   // k=2 -> 16
                fk = 2.0f * x * fkm1 - fkm2; fkm2 = fkm1; fkm1 = fk;
                atomicAdd(&dstF[17], (fk + 1.0f) * hfc);                   // k=3 -> 17
                fk = 2.0f * x * fkm1 - fkm2; fkm2 = fkm1; fkm1 = fk;
                atomicAdd(&dstF[2],  (fk + 1.0f) * hfc);                   // k=4 -> 2
                fk = 2.0f * x * fkm1 - fkm2; fkm2 = fkm1; fkm1 = fk;
                atomicAdd(&dstF[3],  (fk + 1.0f) * hfc);                   // k=5 -> 3
                fk = 2.0f * x * fkm1 - fkm2; fkm2 = fkm1; fkm1 = fk;
                atomicAdd(&dstF[18], (fk + 1.0f) * hfc);                   // k=6 -> 18
                fk = 2.0f * x * fkm1 - fkm2;
                atomicAdd(&dstF[19], (fk + 1.0f) * hfc);                   // k=7 -> 19
            }
        }
    }
    __syncthreads();

    // ---- Phase 2: WMMA contraction, one 16-atom tile per wave ----
    const int lane  = tid & 31;
    const int wave  = tid >> 5;
    const int n     = lane & 15;        // N index (descriptor column)
    const int khalf = lane >> 4;        // 0: K slots {0,1}, 1: K slots {2,3}
    const int aLocM = wave * 16 + n;    // M index = atom within tile
    const int aGlbM = atomBase + aLocM;

    const int taIdx = (aGlbM < N_ATOMS) ? aGlbM : 0;
    const int taRaw = types[taIdx];
    const int ta    = (aGlbM < N_ATOMS) ? taRaw : -1;   // never matches -> zero rows

    // Preload this lane's 16 F values (contiguous thanks to the swizzle):
    // fA[2*p + low] = F-value for K-pair p, element low, this lane-half.
    float fA[16];
    {
        const float* fbase = &F[aLocM * 32 + khalf * 16];
        #pragma unroll
        for (int q = 0; q < 4; ++q) {
            const float4 v = *reinterpret_cast<const float4*>(fbase + 4 * q);
            fA[4 * q + 0] = v.x; fA[4 * q + 1] = v.y;
            fA[4 * q + 2] = v.z; fA[4 * q + 3] = v.w;
        }
    }

    // Batch-preload all 32 B fragments (row K -> (ta_c, tj, k), col n -> d).
    v2f bfrag[32];
    #pragma unroll
    for (int kk = 0; kk < 32; ++kk) {
        const int K  = kk * 4 + khalf * 2;               // even; K,K+1 share (ta_c,tj)
        const int cb = (((K >> 5) * N_TYPES + ((K >> 3) & 3)) * N_DESC + (n & 7)) * K_MAX
                       + (K & 7);
        const float bx = ctab[cb];
        const float by = ctab[cb + 1];
        bfrag[kk].x = (n < N_DESC) ? bx : 0.0f;          // pad columns 8..15 with zero
        bfrag[kk].y = (n < N_DESC) ? by : 0.0f;
    }

    v8f acc = {};
    #pragma unroll
    for (int kk = 0; kk < 32; ++kk) {                    // K = 128, 4 per step
        const bool match = (ta == (kk >> 3));            // center-type block gate
        v2f a;
        a.x = match ? fA[2 * (kk & 7) + 0] : 0.0f;       // register cndmask only
        a.y = match ? fA[2 * (kk & 7) + 1] : 0.0f;
        acc = __builtin_amdgcn_wmma_f32_16x16x4_f32(
                  /*neg_a=*/false, a, /*neg_b=*/false, bfrag[kk],
                  /*c_mod=*/(short)0, acc, /*reuse_a=*/false, /*reuse_b=*/false);
    }

    // ---- store D: VGPR r holds M=r (lanes 0-15) and M=r+8 (lanes 16-31) ----
    if (n < N_DESC) {
        const int moff = khalf * 8;
        #pragma unroll
        for (int r = 0; r < 8; ++r) {
            const int ag = atomBase + wave * 16 + r + moff;
            if (ag < N_ATOMS) out[ag * N_DESC + n] = acc[r];
        }
    }
}

extern "C" void kernel_launch(void* const* d_in, const int* in_sizes, int n_in,
                              void* d_out, int out_size, void* d_ws, size_t ws_size,
                              hipStream_t stream) {
    (void)in_sizes; (void)n_in; (void)d_ws; (void)ws_size; (void)out_size;
    const int*   types     = (const int*)  d_in[0];
    const float* positions = (const float*)d_in[1];
    const int*   nbrs      = (const int*)  d_in[2];
    const float* offsets   = (const float*)d_in[3];
    const float* ctab      = (const float*)d_in[4];
    float*       out       = (float*)d_out;

    const int nblocks = (N_ATOMS + ATOMS_PER_BLOCK - 1) / ATOMS_PER_BLOCK;
    hipLaunchKernelGGL(radial_desc_wmma_kernel, dim3(nblocks), dim3(THREADS), 0, stream,
                       types, positions, nbrs, offsets, ctab, out);
}